// InferenceAndGeneration_64132451664668
// MI455X (gfx1250) — compile-verified
//
#include <hip/hip_runtime.h>

// ---------------------------------------------------------------------------
// Bounding-box-of-mask reduction over 512 images of 256x256 f32.
// Bandwidth bound: 128 MiB read once -> >= ~5.7us at 23.3 TB/s.
// CDNA5 path: async global->LDS (ASYNCcnt) streaming pipeline, depth 8
// (16 waves x 7 outstanding x 512B ~= 57 KB in flight per WGP, sized to
//  cover HBM latency at the per-WGP bandwidth share).
// ---------------------------------------------------------------------------

#ifndef __has_builtin
#define __has_builtin(x) 0
#endif

#if __has_builtin(__builtin_amdgcn_global_load_async_to_lds_b128) && \
    __has_builtin(__builtin_amdgcn_s_wait_asynccnt)
#define USE_ASYNC_LDS 1
#else
#define USE_ASYNC_LDS 0
#endif

#define THREADS 512
#define IMG_ELEMS 65536                     // 256*256
#define ITERS (IMG_ELEMS / (THREADS * 4))   // 32 iterations of float4/thread
#define DEPTH 8                             // async pipeline depth

__device__ __forceinline__ int imin(int a, int b) { return a < b ? a : b; }
__device__ __forceinline__ int imax(int a, int b) { return a > b ? a : b; }

#if USE_ASYNC_LDS
typedef int v4i __attribute__((ext_vector_type(4)));
typedef __attribute__((address_space(1))) v4i gv4i_t;   // global 16B vector
typedef __attribute__((address_space(3))) v4i lv4i_t;   // LDS    16B vector

__device__ __forceinline__ void async_copy16(const float* g, void* lds) {
    float* gnc = const_cast<float*>(g);
    __builtin_amdgcn_global_load_async_to_lds_b128(
        (gv4i_t*)gnc,          // src: global, AS1, v4i
        (lv4i_t*)lds,          // dst: LDS,    AS3, v4i (generic->AS3 cast)
        0, 0);                 // imm offset, imm cpol
}
#endif

__global__ __launch_bounds__(THREADS) void bbox_kernel(
    const float* __restrict__ mixing,
    const int* __restrict__ pad_p,
    const int* __restrict__ minb_p,
    const int* __restrict__ maxb_p,
    float* __restrict__ out,
    int nimg)
{
    const int img = blockIdx.x;
    const int t   = threadIdx.x;
    const float* base = mixing + (size_t)img * IMG_ELEMS;

    const int hbase = (t & 63) << 2;   // this thread's fixed h quad
    const int wlane = t >> 6;          // 0..7 sub-row within a 2048-float chunk

    int minw = 256, maxw = -1;
    unsigned anyj = 0u;                // 4 bits, one per h in the quad

#if USE_ASYNC_LDS
    __shared__ float4 stage[DEPTH][THREADS];

    // Prologue: fill the pipeline (DEPTH-1 chunks in flight).
    #pragma unroll
    for (int p = 0; p < DEPTH - 1; ++p)
        async_copy16(base + p * (THREADS * 4) + 4 * t, &stage[p][t]);

    for (int i = 0; i < ITERS; ++i) {
        const int nx = i + DEPTH - 1;
        if (nx < ITERS) {
            async_copy16(base + nx * (THREADS * 4) + 4 * t,
                         &stage[nx % DEPTH][t]);
            __builtin_amdgcn_s_wait_asynccnt(DEPTH - 1);  // buffer i complete
        } else {
            // Tail: drain everything once; subsequent waits are no-ops.
            __builtin_amdgcn_s_wait_asynccnt(0);
        }
        // Each lane reads back only its own staged quad: no barrier needed.
        const float4 v = stage[i % DEPTH][t];

        unsigned m = (v.x > 0.5f ? 1u : 0u) | (v.y > 0.5f ? 2u : 0u) |
                     (v.z > 0.5f ? 4u : 0u) | (v.w > 0.5f ? 8u : 0u);
        if (m) {
            const int w = i * 8 + wlane;
            minw = imin(minw, w);
            maxw = imax(maxw, w);
            anyj |= m;
        }
    }
#else
    for (int i = 0; i < ITERS; ++i) {
        if (i + 8 < ITERS)
            __builtin_prefetch(base + (i + 8) * (THREADS * 4) + 4 * t, 0, 0);
        const float4 v = *(const float4*)(base + i * (THREADS * 4) + 4 * t);
        unsigned m = (v.x > 0.5f ? 1u : 0u) | (v.y > 0.5f ? 2u : 0u) |
                     (v.z > 0.5f ? 4u : 0u) | (v.w > 0.5f ? 8u : 0u);
        if (m) {
            const int w = i * 8 + wlane;
            minw = imin(minw, w);
            maxw = imax(maxw, w);
            anyj |= m;
        }
    }
#endif

    // Per-thread h extremes from the 4-bit any mask.
    int minh = 256, maxh = -1;
    if (anyj) {
        minh = hbase + (__ffs(anyj) - 1);
        maxh = hbase + (31 - __clz((int)anyj));
    }

    // Wave32 butterfly reduction (warpSize == 32 on gfx1250).
    #pragma unroll
    for (int off = 16; off > 0; off >>= 1) {
        minw = imin(minw, __shfl_xor(minw, off, 32));
        maxw = imax(maxw, __shfl_xor(maxw, off, 32));
        minh = imin(minh, __shfl_xor(minh, off, 32));
        maxh = imax(maxh, __shfl_xor(maxh, off, 32));
    }

    // Cross-wave reduction through LDS (16 waves).
    __shared__ int red[4][THREADS / 32];
    const int wave = t >> 5;
    if ((t & 31) == 0) {
        red[0][wave] = minw;
        red[1][wave] = maxw;
        red[2][wave] = minh;
        red[3][wave] = maxh;
    }
    __syncthreads();

    if (t == 0) {
        #pragma unroll
        for (int w8 = 1; w8 < THREADS / 32; ++w8) {
            minw = imin(minw, red[0][w8]);
            maxw = imax(maxw, red[1][w8]);
            minh = imin(minh, red[2][w8]);
            maxh = imax(maxh, red[3][w8]);
        }

        const float Wf  = 256.0f, Hf = 256.0f;
        const float pad = (float)(*pad_p);
        const float mnb = (float)(*minb_p);
        const float mxb = (float)(*maxb_p);

        // Reproduce reference semantics exactly (incl. empty mask):
        //   ideal_x1 = any ? minw : W   (minw init 256)
        //   ideal_x3 = max(maxw, 0)     (maxw init -1)
        float ix1 = (float)minw;
        float ix3 = fmaxf((float)maxw, 0.0f);
        float iy1 = (float)minh;
        float iy3 = fmaxf((float)maxh, 0.0f);

        ix1 = fminf(fmaxf(ix1 - pad, 0.0f), Wf);
        iy1 = fminf(fmaxf(iy1 - pad, 0.0f), Hf);
        ix3 = fminf(fmaxf(ix3 + pad, 0.0f), Wf);
        iy3 = fminf(fmaxf(iy3 + pad, 0.0f), Hf);

        const float bx = 0.5f * (ix1 + ix3);
        const float by = 0.5f * (iy1 + iy3);
        const float bw = fminf(fmaxf(ix3 - ix1, mnb), mxb);
        const float bh = fminf(fmaxf(iy3 - iy1, mnb), mxb);

        out[0 * nimg + img] = bx;
        out[1 * nimg + img] = by;
        out[2 * nimg + img] = bw;
        out[3 * nimg + img] = bh;
    }
}

extern "C" void kernel_launch(void* const* d_in, const int* in_sizes, int n_in,
                              void* d_out, int out_size, void* d_ws, size_t ws_size,
                              hipStream_t stream) {
    const float* mixing = (const float*)d_in[0];
    const int*   pad_p  = (const int*)d_in[1];
    const int*   minb_p = (const int*)d_in[2];
    const int*   maxb_p = (const int*)d_in[3];
    float*       out    = (float*)d_out;

    const int nimg = in_sizes[0] / IMG_ELEMS;   // K*B = 512

    bbox_kernel<<<nimg, THREADS, 0, stream>>>(mixing, pad_p, minb_p, maxb_p,
                                              out, nimg);
}